// DynamicStateMixer_51539608276
// MI455X (gfx1250) — compile-verified
//
#include <hip/hip_runtime.h>

// ---------------- problem constants (from reference) ----------------
#define B_    8
#define L_    24
#define HPL_  32
#define HW_   4096      // 64*64
#define NCH_  768       // N = L*HPL
#define HID_  10        // max(8, 32//3)

// d_out layout (floats), outputs concatenated in return order
#define LOGITS_OFF 0UL
#define COEFF_OFF  50331648UL     // B*L*32*2*HW
#define WA_OFF     100663296UL
#define WB_OFF     125829120UL
#define MIXED_OFF  150994944UL

typedef __attribute__((ext_vector_type(2))) float v2f;
typedef __attribute__((ext_vector_type(8))) float v8f;

// ---------------- kernel 1: per-channel means of x (B*1536 channels) ----------------
__global__ __launch_bounds__(256) void xmean_kernel(const float* __restrict__ x,
                                                    float* __restrict__ xmean) {
  __shared__ float red[256];
  const float* src = x + (size_t)blockIdx.x * HW_;
  float s = 0.0f;
  for (int i = threadIdx.x; i < HW_; i += 256) s += src[i];
  red[threadIdx.x] = s;
  __syncthreads();
  for (int off = 128; off > 0; off >>= 1) {
    if (threadIdx.x < off) red[threadIdx.x] += red[threadIdx.x + off];
    __syncthreads();
  }
  if (threadIdx.x == 0) xmean[blockIdx.x] = red[0] * (1.0f / HW_);
}

// ---------------- kernel 2: per-channel means of mixed(layer l) -> prevmean ----------------
__global__ __launch_bounds__(256) void layer_mean_kernel(const float* __restrict__ mixed,
                                                         float* __restrict__ prevmean,
                                                         int l) {
  __shared__ float red[256];
  int b = blockIdx.x >> 5;        // /32
  int c = blockIdx.x & 31;
  const float* src = mixed + ((size_t)b * NCH_ + (size_t)l * HPL_ + c) * HW_;
  float s = 0.0f;
  for (int i = threadIdx.x; i < HW_; i += 256) s += src[i];
  red[threadIdx.x] = s;
  __syncthreads();
  for (int off = 128; off > 0; off >>= 1) {
    if (threadIdx.x < off) red[threadIdx.x] += red[threadIdx.x + off];
    __syncthreads();
  }
  if (threadIdx.x == 0) prevmean[blockIdx.x] = red[0] * (1.0f / HW_);
}

// ---------------- kernel 3: gating MLP  gc(96) -> SiLU(10) -> gl(64), all batches ----------------
__global__ __launch_bounds__(768) void gating_kernel(const float* __restrict__ xmean,
                                                     const float* __restrict__ prevmean,
                                                     const float* __restrict__ W1,
                                                     const float* __restrict__ b1,
                                                     const float* __restrict__ W2,
                                                     const float* __restrict__ b2,
                                                     float* __restrict__ gl,
                                                     int l) {
  __shared__ float s_gc[B_][96];
  __shared__ float s_h[B_][HID_];
  int t = threadIdx.x;
  {
    int b = t / 96, k = t % 96;      // t < 768 always
    float v;
    if (k < HPL_)       v = xmean[b * (2 * NCH_) + l * HPL_ + k];
    else if (k < 64)    v = xmean[b * (2 * NCH_) + NCH_ + l * HPL_ + (k - HPL_)];
    else                v = (l == 0) ? 0.0f : prevmean[b * HPL_ + (k - 64)];
    s_gc[b][k] = v;
  }
  __syncthreads();
  if (t < B_ * HID_) {
    int b = t / HID_, h = t % HID_;
    const float* w = W1 + ((size_t)l * HID_ + h) * 96;
    float s = b1[l * HID_ + h];
    #pragma unroll 8
    for (int k = 0; k < 96; ++k) s += s_gc[b][k] * w[k];
    s_h[b][h] = s / (1.0f + __expf(-s));   // SiLU
  }
  __syncthreads();
  if (t < B_ * 64) {
    int b = t / 64, o = t % 64;
    const float* w = W2 + ((size_t)l * 64 + o) * HID_;
    float s = b2[l * 64 + o];
    #pragma unroll
    for (int h = 0; h < HID_; ++h) s += s_h[b][h] * w[h];
    gl[b * 64 + o] = s;
  }
}

// Epilogue-only cat pick: xa/xb/pv are per-lane column pointers (pixel p folded in).
template <bool FIRST>
__device__ __forceinline__ float cat_pick(const float* __restrict__ xa,
                                          const float* __restrict__ xb,
                                          const float* __restrict__ pv, int ch) {
  if (FIRST) {
    if (ch >= 64) return 0.0f;          // prev == zeros on layer 0
    const float* q = (ch < HPL_) ? (xa + (size_t)ch * HW_)
                                 : (xb + (size_t)(ch - HPL_) * HW_);
    return *q;
  } else {
    const float* q = (ch < HPL_) ? (xa + (size_t)ch * HW_)
                   : (ch < 64)   ? (xb + (size_t)(ch - HPL_) * HW_)
                                 : (pv + (size_t)(ch - 64) * HW_);
    return *q;
  }
}

// ---------------- kernel 4: WMMA einsum + mixing epilogue ----------------
// grid: (HW/128, B); block: 256 threads = 8 waves.
// Each wave: sl(64 x 16 px) = Wc_l(64 x 96) @ cat(96 x 16), K tiled by 4 via
// v_wmma_f32_16x16x4_f32; K loop split into 3 branch-free region loops.
template <bool FIRST>
__global__ __launch_bounds__(256) void mix_kernel(const float* __restrict__ x,
                                                  const float* __restrict__ Wc,
                                                  const float* __restrict__ bc,
                                                  const float* __restrict__ gl,
                                                  float* __restrict__ out,
                                                  int l) {
  __shared__ float s_gbc[64];           // gl[b,o] + bc[l,o]
  const int b    = blockIdx.y;
  if (threadIdx.x < 64)
    s_gbc[threadIdx.x] = gl[b * 64 + threadIdx.x] + bc[l * 64 + threadIdx.x];
  __syncthreads();

  const int wave = threadIdx.x >> 5;
  const int lane = threadIdx.x & 31;
  const int hi   = lane >> 4;          // half-wave select
  const int n    = lane & 15;          // N index (pixel) / M row within tile
  const int p    = blockIdx.x * 128 + wave * 16 + n;

  const float* WcL   = Wc + (size_t)l * 64 * 96;
  // per-lane column pointers (pixel folded in)
  const float* xaCol = x + ((size_t)b * (2 * NCH_) + (size_t)l * HPL_) * HW_ + p;
  const float* xbCol = xaCol + (size_t)NCH_ * HW_;
  const float* pvCol = FIRST ? xaCol   // unused when FIRST (kept valid for codegen)
                             : out + MIXED_OFF + ((size_t)b * NCH_ + (size_t)(l - 1) * HPL_) * HW_ + p;

  v8f acc[4] = {v8f{}, v8f{}, v8f{}, v8f{}};

  // ---- region 0: channels 0..31 from xa ----
  #pragma unroll
  for (int k0 = 0; k0 < 32; k0 += 4) {
    const int kc = k0 + 2 * hi;                       // {kc, kc+1} K pair for this half-wave
    v2f bf;
    bf.x = xaCol[(size_t)kc * HW_];
    bf.y = xaCol[(size_t)kc * HW_ + HW_];
    #pragma unroll
    for (int mt = 0; mt < 4; ++mt) {
      const float* a = WcL + (size_t)(mt * 16 + n) * 96 + kc;
      v2f af; af.x = a[0]; af.y = a[1];
      acc[mt] = __builtin_amdgcn_wmma_f32_16x16x4_f32(
          false, af, false, bf, (short)0, acc[mt], false, false);
    }
  }
  // ---- region 1: channels 32..63 from xb ----
  #pragma unroll
  for (int k0 = 0; k0 < 32; k0 += 4) {
    const int kc = k0 + 2 * hi;
    v2f bf;
    bf.x = xbCol[(size_t)kc * HW_];
    bf.y = xbCol[(size_t)kc * HW_ + HW_];
    #pragma unroll
    for (int mt = 0; mt < 4; ++mt) {
      const float* a = WcL + (size_t)(mt * 16 + n) * 96 + 32 + kc;
      v2f af; af.x = a[0]; af.y = a[1];
      acc[mt] = __builtin_amdgcn_wmma_f32_16x16x4_f32(
          false, af, false, bf, (short)0, acc[mt], false, false);
    }
  }
  // ---- region 2: channels 64..95 from prev (skipped entirely on layer 0) ----
  if (!FIRST) {
    #pragma unroll
    for (int k0 = 0; k0 < 32; k0 += 4) {
      const int kc = k0 + 2 * hi;
      v2f bf;
      bf.x = pvCol[(size_t)kc * HW_];
      bf.y = pvCol[(size_t)kc * HW_ + HW_];
      #pragma unroll
      for (int mt = 0; mt < 4; ++mt) {
        const float* a = WcL + (size_t)(mt * 16 + n) * 96 + 64 + kc;
        v2f af; af.x = a[0]; af.y = a[1];
        acc[mt] = __builtin_amdgcn_wmma_f32_16x16x4_f32(
            false, af, false, bf, (short)0, acc[mt], false, false);
      }
    }
  }

  // ---- epilogue ----
  // D layout: VGPR r -> M = r + 8*hi (within 16-row tile), N = lane&15.
  float* logitsP = out + LOGITS_OFF;
  float* coeffP  = out + COEFF_OFF;
  float* waP     = out + WA_OFF;
  float* wbP     = out + WB_OFF;
  float* mixedP  = out + MIXED_OFF;

  #pragma unroll
  for (int mt = 0; mt < 4; ++mt) {
    #pragma unroll
    for (int t = 0; t < 4; ++t) {
      const int r0 = 2 * t;
      const int o0 = mt * 16 + 8 * hi + r0;   // even output channel
      const int c  = o0 >> 1;                 // half-channel index
      float lg0 = acc[mt][r0]     + s_gbc[o0];
      float lg1 = acc[mt][r0 + 1] + s_gbc[o0 + 1];
      float wbv = 1.0f / (1.0f + __expf(lg0 - lg1));  // sigmoid(lg1 - lg0)
      float wav = 1.0f - wbv;
      // xg[:,c,0] = cat[3c], xg[:,c,1] = cat[3c+1]
      float xaa = cat_pick<FIRST>(xaCol, xbCol, pvCol, 3 * c);
      float xbb = cat_pick<FIRST>(xaCol, xbCol, pvCol, 3 * c + 1);
      float mx  = xaa + wbv * (xbb - xaa);

      size_t base2 = ((((size_t)b * L_ + l) * HPL_ + c) * 2) * HW_ + p;
      __builtin_nontemporal_store(lg0, logitsP + base2);
      __builtin_nontemporal_store(lg1, logitsP + base2 + HW_);
      __builtin_nontemporal_store(wav, coeffP + base2);
      __builtin_nontemporal_store(wbv, coeffP + base2 + HW_);
      size_t base1 = (((size_t)b * L_ + l) * HPL_ + c) * HW_ + p;
      __builtin_nontemporal_store(wav, waP + base1);
      __builtin_nontemporal_store(wbv, wbP + base1);
      // mixed is re-read by layer l+1 -> keep cacheable (L2-resident, 4 MB/layer)
      mixedP[((size_t)b * NCH_ + (size_t)l * HPL_ + c) * HW_ + p] = mx;
    }
  }
}

// ---------------- host launcher ----------------
extern "C" void kernel_launch(void* const* d_in, const int* in_sizes, int n_in,
                              void* d_out, int out_size, void* d_ws, size_t ws_size,
                              hipStream_t stream) {
  const float* x  = (const float*)d_in[0];
  const float* W1 = (const float*)d_in[1];
  const float* b1 = (const float*)d_in[2];
  const float* W2 = (const float*)d_in[3];
  const float* b2 = (const float*)d_in[4];
  const float* Wc = (const float*)d_in[5];
  const float* bc = (const float*)d_in[6];
  float* out = (float*)d_out;
  float* ws  = (float*)d_ws;

  float* xmean    = ws;                       // B * 1536
  float* prevmean = xmean + B_ * 2 * NCH_;    // B * 32
  float* gl       = prevmean + B_ * HPL_;     // B * 64

  xmean_kernel<<<B_ * 2 * NCH_, 256, 0, stream>>>(x, xmean);

  for (int l = 0; l < L_; ++l) {
    gating_kernel<<<1, 768, 0, stream>>>(xmean, prevmean, W1, b1, W2, b2, gl, l);
    if (l == 0)
      mix_kernel<true><<<dim3(HW_ / 128, B_), 256, 0, stream>>>(x, Wc, bc, gl, out, l);
    else
      mix_kernel<false><<<dim3(HW_ / 128, B_), 256, 0, stream>>>(x, Wc, bc, gl, out, l);
    if (l + 1 < L_)
      layer_mean_kernel<<<B_ * HPL_, 256, 0, stream>>>(out + MIXED_OFF, prevmean, l);
  }
}